// NodeGraphNeighbourhood_7060926234625
// MI455X (gfx1250) — compile-verified
//
#include <hip/hip_runtime.h>

// Graph depth-1 neighbourhood mean + concat.
//   in0: node_tensor  [N, 128] f32
//   in1: edge_src     [E] int
//   in2: edge_dst     [E] int
//   out: [N, 256] f32 : out[:, :128] = node, out[:, 128:] = (seg_sum + node)/(deg+1)
//
// Memory-bound irregular gather/scatter (~0.12 FLOP/byte): no matrix structure,
// so no WMMA. Optimized for MI455X by keeping both the node tensor (51.2MB) and
// the f32 accumulator (51.2MB) L2-resident (192MB L2), using native
// global_atomic_add_f32 (unsafeAtomicAdd), and global_prefetch_b8 to hide the
// gather latency. Edge indices are scalarized via readfirstlane so the hot loop
// uses SGPR base + lane-offset addressing.

#define D_FEAT 128
#define ROW_OUT 256

__global__ void ngn_init_kernel(float* __restrict__ out, float* __restrict__ deg,
                                int n_nodes) {
    int idx = blockIdx.x * blockDim.x + threadIdx.x;
    int total = n_nodes * (D_FEAT / 4);           // 32 float4 chunks per node
    if (idx < total) {
        int node  = idx >> 5;
        int chunk = idx & 31;
        float4 z = make_float4(0.f, 0.f, 0.f, 0.f);
        *(float4*)(out + (size_t)node * ROW_OUT + D_FEAT + chunk * 4) = z;
        if (chunk == 0) deg[node] = 0.f;
    }
}

__global__ void ngn_edge_kernel(const float* __restrict__ node,
                                const int* __restrict__ esrc,
                                const int* __restrict__ edst,
                                float* __restrict__ out,
                                float* __restrict__ deg,
                                int n_edges) {
    const int lane   = threadIdx.x & 31;
    const int wave   = blockIdx.x * (blockDim.x >> 5) + (threadIdx.x >> 5);
    const int nwaves = gridDim.x * (blockDim.x >> 5);

    for (int e = wave; e < n_edges; e += nwaves) {
        // e is wave-uniform: force indices into SGPRs so row bases are scalar.
        const int s = __builtin_amdgcn_readfirstlane(esrc[e]);
        const int d = __builtin_amdgcn_readfirstlane(edst[e]);

        // Prefetch next iteration's source row (gfx1250 global_prefetch_b8).
        const int en = e + nwaves;
        if (en < n_edges) {
            const int sn = __builtin_amdgcn_readfirstlane(esrc[en]);
            __builtin_prefetch(node + (size_t)sn * D_FEAT + lane * 4, 0, 3);
        }

        // Coalesced 512B row gather: one float4 per lane (global_load_b128).
        const float4 v = *(const float4*)(node + (size_t)s * D_FEAT + lane * 4);

        // Scatter-add into the "reduced" half of the output row (L2 f32 atomics).
        float* a = out + (size_t)d * ROW_OUT + D_FEAT + lane * 4;
        unsafeAtomicAdd(a + 0, v.x);
        unsafeAtomicAdd(a + 1, v.y);
        unsafeAtomicAdd(a + 2, v.z);
        unsafeAtomicAdd(a + 3, v.w);

        if (lane == 0) unsafeAtomicAdd(deg + d, 1.0f);
    }
}

__global__ void ngn_final_kernel(const float* __restrict__ node,
                                 const float* __restrict__ deg,
                                 float* __restrict__ out,
                                 int n_nodes) {
    int idx = blockIdx.x * blockDim.x + threadIdx.x;
    int total = n_nodes * (D_FEAT / 4);
    if (idx < total) {
        int i     = idx >> 5;                     // node (uniform across a wave)
        int chunk = idx & 31;                     // float4 chunk within row

        const float4 nv = *(const float4*)(node + (size_t)i * D_FEAT + chunk * 4);
        float* accp = out + (size_t)i * ROW_OUT + D_FEAT + chunk * 4;
        const float4 ac = *(const float4*)accp;
        const float inv = 1.0f / (deg[i] + 1.0f);

        float4 r;
        r.x = (ac.x + nv.x) * inv;
        r.y = (ac.y + nv.y) * inv;
        r.z = (ac.z + nv.z) * inv;
        r.w = (ac.w + nv.w) * inv;

        *(float4*)(out + (size_t)i * ROW_OUT + chunk * 4) = nv;   // concat: self features
        *(float4*)accp = r;                                       // concat: mean reduce
    }
}

extern "C" void kernel_launch(void* const* d_in, const int* in_sizes, int n_in,
                              void* d_out, int out_size, void* d_ws, size_t ws_size,
                              hipStream_t stream) {
    const float* node = (const float*)d_in[0];
    const int*   esrc = (const int*)d_in[1];
    const int*   edst = (const int*)d_in[2];
    float*       out  = (float*)d_out;
    float*       deg  = (float*)d_ws;            // n_nodes floats of scratch

    const int n_nodes = in_sizes[0] / D_FEAT;
    const int n_edges = in_sizes[1];

    // K1: zero accumulator half of out + degree array.
    {
        int total  = n_nodes * (D_FEAT / 4);
        int blocks = (total + 255) / 256;
        ngn_init_kernel<<<blocks, 256, 0, stream>>>(out, deg, n_nodes);
    }

    // K2: wave-per-edge gather + scatter-add (grid-stride, prefetched).
    {
        int blocks = 2560;                        // 20480 waves in flight
        ngn_edge_kernel<<<blocks, 256, 0, stream>>>(node, esrc, edst, out, deg, n_edges);
    }

    // K3: finalize mean (+self) and write concatenated output.
    {
        int total  = n_nodes * (D_FEAT / 4);
        int blocks = (total + 255) / 256;
        ngn_final_kernel<<<blocks, 256, 0, stream>>>(node, deg, out, n_nodes);
    }
}